// GATv2Encoder_80041010528711
// MI455X (gfx1250) — compile-verified
//
#include <hip/hip_runtime.h>
#include <math.h>

typedef float v2f __attribute__((ext_vector_type(2)));
typedef float v8f __attribute__((ext_vector_type(8)));

#define D 128

// ---------- helpers ----------

__device__ __forceinline__ v8f wmma4(v2f a, v2f b, v8f c) {
  // V_WMMA_F32_16X16X4_F32 : D = A(16x4) * B(4x16) + C(16x16), fp32 exact
  return __builtin_amdgcn_wmma_f32_16x16x4_f32(false, a, false, b, (short)0, c,
                                               false, false);
}

__device__ __forceinline__ float gelu_exact(float x) {
  return 0.5f * x * (1.0f + erff(x * 0.70710678118654752440f));
}

// order-preserving float <-> uint for atomicMax on floats
__device__ __forceinline__ unsigned fenc(float f) {
  unsigned u = __float_as_uint(f);
  return (u & 0x80000000u) ? ~u : (u | 0x80000000u);
}
__device__ __forceinline__ float fdec(unsigned u) {
  return __uint_as_float((u & 0x80000000u) ? (u ^ 0x80000000u) : ~u);
}
#define AMAX_INIT 0x007FFFFFu  /* fenc(-inf) */

// ---------- edge sort by relation (counting sort) ----------

__global__ void k_zero8(int* counts) {
  if (threadIdx.x < 8) counts[threadIdx.x] = 0;
}

__global__ void k_hist(const int* __restrict__ et, int* __restrict__ counts, int E) {
  int i = blockIdx.x * blockDim.x + threadIdx.x;
  if (i < E) atomicAdd(&counts[et[i]], 1);
}

__global__ void k_scan(const int* __restrict__ counts, int* __restrict__ cursor, int R) {
  if (threadIdx.x == 0 && blockIdx.x == 0) {
    int run = 0;
    for (int r = 0; r < R; ++r) { cursor[r] = run; run += counts[r]; }
  }
}

__global__ void k_scatter(const int* __restrict__ src, const int* __restrict__ dst,
                          const int* __restrict__ et, int* __restrict__ cursor,
                          int* __restrict__ srcS, int* __restrict__ dstS,
                          int* __restrict__ etS, int E) {
  int i = blockIdx.x * blockDim.x + threadIdx.x;
  if (i < E) {
    int r = et[i];
    int pos = atomicAdd(&cursor[r], 1);
    srcS[pos] = src[i];
    dstS[pos] = dst[i];
    etS[pos]  = r;
  }
}

// ---------- per-layer init ----------

__global__ void k_init(float* __restrict__ accum, float* __restrict__ denom,
                       unsigned* __restrict__ amaxE, int n) {
  int idx = blockIdx.x * blockDim.x + threadIdx.x;
  if (idx >= n * D) return;
  accum[idx] = 0.0f;
  if (idx < n * 4) { denom[idx] = 0.0f; amaxE[idx] = AMAX_INIT; }
}

// ---------- xl = x @ Wl + bl  (WMMA, 16 rows x 128 cols per block) ----------

__global__ void k_node_gemm(const float* __restrict__ X, const float* __restrict__ W,
                            const float* __restrict__ bias, float* __restrict__ Y) {
  __shared__ float As[16][132];
  int t = threadIdx.x;
  int row0 = blockIdx.x * 16;
  #pragma unroll
  for (int j = 0; j < 8; ++j) {
    int lin = t + j * 256;
    int r = lin >> 7, c = lin & 127;
    As[r][c] = X[(size_t)(row0 + r) * D + c];
  }
  __syncthreads();
  int lane = t & 31, wave = t >> 5;
  int lN = lane & 15, lH = lane >> 4;
  int col = wave * 16 + lN;
  v8f acc = {};
  #pragma unroll 4
  for (int k = 0; k < 32; ++k) {
    v2f a = *(const v2f*)&As[lN][4 * k + 2 * lH];
    v2f b;
    b.x = W[(4 * k + 2 * lH) * D + col];
    b.y = W[(4 * k + 2 * lH + 1) * D + col];
    acc = wmma4(a, b, acc);
  }
  float bv = bias[col];
  #pragma unroll
  for (int v = 0; v < 8; ++v)
    Y[(size_t)(row0 + v + 8 * lH) * D + col] = acc[v] + bv;
}

// ---------- fused edge kernel: rel-GEMM + gelu + We-GEMM + attention logits ----
// 32 edges per block; each wave owns a 16-col slice and TWO 16-row tiles, so
// every B-fragment load feeds two WMMAs (halves L2 B-traffic).

__global__ void k_edge_pass1(const float* __restrict__ X, const float* __restrict__ Rm,
                             const float* __restrict__ We, const float* __restrict__ xl,
                             const float* __restrict__ attf,
                             const int* __restrict__ srcS, const int* __restrict__ dstS,
                             const int* __restrict__ etS,
                             float* __restrict__ alphaS, unsigned* __restrict__ amaxE) {
  __shared__ float As[32][260];   // cat(x[src], x[dst])  32 x 256
  __shared__ float Ts[32][132];   // gelu(A @ R)          32 x 128
  __shared__ float Su[32][132];   // xl[src]+xl[dst]      32 x 128
  __shared__ float al_s[128];     // alpha partials       32 x 4
  __shared__ int   ets[32];

  int t = threadIdx.x;
  int e0 = blockIdx.x * 32;

  #pragma unroll
  for (int j = 0; j < 32; ++j) {
    int lin = t + j * 256;
    int r = lin >> 8, c = lin & 255;
    int e = e0 + r;
    int node = (c < 128) ? srcS[e] : dstS[e];
    As[r][c] = X[(size_t)node * D + (c & 127)];
  }
  #pragma unroll
  for (int j = 0; j < 16; ++j) {
    int lin = t + j * 256;
    int r = lin >> 7, c = lin & 127;
    int e = e0 + r;
    Su[r][c] = xl[(size_t)srcS[e] * D + c] + xl[(size_t)dstS[e] * D + c];
  }
  if (t < 128) al_s[t] = 0.0f;
  if (t < 32) ets[t] = etS[e0 + t];
  __syncthreads();

  int lane = t & 31, wave = t >> 5;
  int lN = lane & 15, lH = lane >> 4;
  int col = wave * 16 + lN;
  int myet0 = ets[lN];
  int myet1 = ets[16 + lN];
  int rmin = ets[0], rmax = ets[0];
  #pragma unroll
  for (int i = 1; i < 32; ++i) { int e = ets[i]; rmin = min(rmin, e); rmax = max(rmax, e); }

  // GEMM1: 32x256 @ 256x128, relation-masked rows (sorted => usually 1 pass)
  v8f acc0 = {}, acc1 = {};
  for (int r = rmin; r <= rmax; ++r) {
    const float* B = Rm + (size_t)r * 256 * D;
    float m0 = (myet0 == r) ? 1.0f : 0.0f;
    float m1 = (myet1 == r) ? 1.0f : 0.0f;
    #pragma unroll 4
    for (int k = 0; k < 64; ++k) {
      v2f b;
      b.x = B[(4 * k + 2 * lH) * D + col];
      b.y = B[(4 * k + 2 * lH + 1) * D + col];
      v2f a0 = *(const v2f*)&As[lN][4 * k + 2 * lH];
      v2f a1 = *(const v2f*)&As[16 + lN][4 * k + 2 * lH];
      a0.x *= m0; a0.y *= m0;
      a1.x *= m1; a1.y *= m1;
      acc0 = wmma4(a0, b, acc0);
      acc1 = wmma4(a1, b, acc1);
    }
  }
  #pragma unroll
  for (int v = 0; v < 8; ++v) {
    Ts[v + 8 * lH][col]      = gelu_exact(acc0[v]);
    Ts[16 + v + 8 * lH][col] = gelu_exact(acc1[v]);
  }
  __syncthreads();

  // GEMM2: 32x128 @ 128x128 (We), B-fragment shared by both row tiles
  v8f c0 = {}, c1 = {};
  #pragma unroll 4
  for (int k = 0; k < 32; ++k) {
    v2f b;
    b.x = We[(4 * k + 2 * lH) * D + col];
    b.y = We[(4 * k + 2 * lH + 1) * D + col];
    v2f a0 = *(const v2f*)&Ts[lN][4 * k + 2 * lH];
    v2f a1 = *(const v2f*)&Ts[16 + lN][4 * k + 2 * lH];
    c0 = wmma4(a0, b, c0);
    c1 = wmma4(a1, b, c1);
  }

  // m = GEMM2 + xl[src]+xl[dst]; alpha = sum_c leaky(m)*att
  float av = attf[col];
  int head = wave >> 1;      // 16-col slice -> half of one 32-col head
  #pragma unroll
  for (int v = 0; v < 8; ++v) {
    #pragma unroll
    for (int tile = 0; tile < 2; ++tile) {
      int row = tile * 16 + v + 8 * lH;
      float m = (tile ? c1[v] : c0[v]) + Su[row][col];
      float lm = (m > 0.0f) ? m : 0.2f * m;
      float p = lm * av;
      p += __shfl_xor(p, 1);
      p += __shfl_xor(p, 2);
      p += __shfl_xor(p, 4);
      p += __shfl_xor(p, 8);
      if (lN == 0) atomicAdd(&al_s[row * 4 + head], p);
    }
  }
  __syncthreads();

  if (t < 128) {
    int r = t >> 2, h = t & 3;
    float al = al_s[t];
    int e = e0 + r;
    alphaS[(size_t)e * 4 + h] = al;
    atomicMax(&amaxE[(size_t)dstS[e] * 4 + h], fenc(al));
  }
}

// ---------- softmax numerator/denominator scatter ----------

__global__ void k_edge_pass2(const int* __restrict__ srcS, const int* __restrict__ dstS,
                             const float* __restrict__ alphaS,
                             const unsigned* __restrict__ amaxE,
                             const float* __restrict__ xl, float* __restrict__ accum,
                             float* __restrict__ denom, int E) {
  long long idx = (long long)blockIdx.x * blockDim.x + threadIdx.x;
  if (idx >= (long long)E * D) return;
  int e  = (int)(idx >> 7);
  int ch = (int)(idx & 127);
  int h  = ch >> 5;
  int s = srcS[e], d = dstS[e];
  float mx = fdec(amaxE[(size_t)d * 4 + h]);
  if (!(mx > -3.0e38f)) mx = 0.0f;          // non-finite max -> 0 (ref semantics)
  float a = expf(alphaS[(size_t)e * 4 + h] - mx);
  unsafeAtomicAdd(&accum[(size_t)d * D + ch], a * xl[(size_t)s * D + ch]);
  if ((ch & 31) == 0) unsafeAtomicAdd(&denom[(size_t)d * 4 + h], a);
}

// ---------- normalize + bias (+gelu for layer 0) ----------

__global__ void k_finalize(const float* __restrict__ accum, const float* __restrict__ denom,
                           const float* __restrict__ bias, float* __restrict__ xout,
                           int n, int dogelu) {
  int idx = blockIdx.x * blockDim.x + threadIdx.x;
  if (idx >= n * D) return;
  int nd = idx >> 7, ch = idx & 127, h = ch >> 5;
  float v = accum[idx] / (denom[nd * 4 + h] + 1e-16f) + bias[ch];
  if (dogelu) v = gelu_exact(v);
  xout[idx] = v;
}

// ---------- output head: Linear (WMMA) + LayerNorm, fused ----------

__global__ void k_head(const float* __restrict__ Xb, const float* __restrict__ W,
                       const float* __restrict__ bo, const float* __restrict__ g,
                       const float* __restrict__ bb, float* __restrict__ out) {
  __shared__ float As[16][132];
  __shared__ float Ys[16][132];
  __shared__ float r1[16][17], r2[16][17];
  __shared__ float muS[16], rsS[16];

  int t = threadIdx.x;
  int row0 = blockIdx.x * 16;
  #pragma unroll
  for (int j = 0; j < 8; ++j) {
    int lin = t + j * 256;
    int r = lin >> 7, c = lin & 127;
    As[r][c] = Xb[(size_t)(row0 + r) * D + c];
  }
  __syncthreads();

  int lane = t & 31, wave = t >> 5;
  int lN = lane & 15, lH = lane >> 4;
  int col = wave * 16 + lN;
  v8f acc = {};
  #pragma unroll 4
  for (int k = 0; k < 32; ++k) {
    v2f a = *(const v2f*)&As[lN][4 * k + 2 * lH];
    v2f b;
    b.x = W[(4 * k + 2 * lH) * D + col];
    b.y = W[(4 * k + 2 * lH + 1) * D + col];
    acc = wmma4(a, b, acc);
  }
  float bv = bo[col];
  #pragma unroll
  for (int v = 0; v < 8; ++v) Ys[v + 8 * lH][col] = acc[v] + bv;
  __syncthreads();

  // LayerNorm over 128 cols per row: 16 partial sums of 8 elems each
  {
    int r = t & 15, p = t >> 4;
    float s1 = 0.0f, s2 = 0.0f;
    #pragma unroll
    for (int c = 0; c < 8; ++c) {
      float y = Ys[r][p * 8 + c];
      s1 += y; s2 += y * y;
    }
    r1[r][p] = s1; r2[r][p] = s2;
  }
  __syncthreads();
  if (t < 16) {
    float s1 = 0.0f, s2 = 0.0f;
    #pragma unroll
    for (int p = 0; p < 16; ++p) { s1 += r1[t][p]; s2 += r2[t][p]; }
    float mu = s1 * (1.0f / 128.0f);
    float var = s2 * (1.0f / 128.0f) - mu * mu;
    muS[t] = mu;
    rsS[t] = rsqrtf(var + 1e-12f);
  }
  __syncthreads();
  #pragma unroll
  for (int j = 0; j < 8; ++j) {
    int lin = t + j * 256;
    int r = lin >> 7, c = lin & 127;
    out[(size_t)(row0 + r) * D + c] = (Ys[r][c] - muS[r]) * rsS[r] * g[c] + bb[c];
  }
}

// ---------- launcher ----------

extern "C" void kernel_launch(void* const* d_in, const int* in_sizes, int n_in,
                              void* d_out, int out_size, void* d_ws, size_t ws_size,
                              hipStream_t stream) {
  const float* embs  = (const float*)d_in[0];
  const int*   eidx  = (const int*)d_in[1];
  const int*   etype = (const int*)d_in[2];
  const float* relm  = (const float*)d_in[3];
  const float* Wl[2]   = {(const float*)d_in[4],  (const float*)d_in[9]};
  const float* bl[2]   = {(const float*)d_in[5],  (const float*)d_in[10]};
  const float* Wed[2]  = {(const float*)d_in[6],  (const float*)d_in[11]};
  const float* att[2]  = {(const float*)d_in[7],  (const float*)d_in[12]};
  const float* bias[2] = {(const float*)d_in[8],  (const float*)d_in[13]};
  const float* outw = (const float*)d_in[14];
  const float* outb = (const float*)d_in[15];
  const float* lng  = (const float*)d_in[16];
  const float* lnb  = (const float*)d_in[17];

  const int N  = in_sizes[0] / D;
  const int E  = in_sizes[2];
  const int R  = in_sizes[3] / (2 * 2 * D * D);
  const int BS = out_size / D;
  const int* src = eidx;
  const int* dst = eidx + E;

  char* wsb = (char*)d_ws;
  size_t off = 0;
  auto carve = [&](size_t bytes) -> void* {
    off = (off + 255) & ~(size_t)255;
    void* p = wsb + off;
    off += bytes;
    return p;
  };
  float*    xl     = (float*)   carve((size_t)N * D * 4);
  float*    xA     = (float*)   carve((size_t)N * D * 4);
  float*    xB     = (float*)   carve((size_t)N * D * 4);
  float*    accum  = (float*)   carve((size_t)N * D * 4);
  float*    alphaS = (float*)   carve((size_t)E * 4 * 4);
  float*    denom  = (float*)   carve((size_t)N * 4 * 4);
  unsigned* amaxE  = (unsigned*)carve((size_t)N * 4 * 4);
  int*      srcS   = (int*)     carve((size_t)E * 4);
  int*      dstS   = (int*)     carve((size_t)E * 4);
  int*      etS    = (int*)     carve((size_t)E * 4);
  int*      counts = (int*)     carve(64);
  int*      cursor = (int*)     carve(64);

  // sort edges by relation (counting sort)
  k_zero8<<<1, 32, 0, stream>>>(counts);
  k_hist<<<(E + 255) / 256, 256, 0, stream>>>(etype, counts, E);
  k_scan<<<1, 1, 0, stream>>>(counts, cursor, R);
  k_scatter<<<(E + 255) / 256, 256, 0, stream>>>(src, dst, etype, cursor,
                                                 srcS, dstS, etS, E);

  for (int l = 0; l < 2; ++l) {
    const float* xin = l ? (const float*)xA : embs;
    const float* Rlayer = relm + (size_t)l * R * 2 * D * D;
    k_init<<<(N * D + 255) / 256, 256, 0, stream>>>(accum, denom, amaxE, N);
    k_node_gemm<<<N / 16, 256, 0, stream>>>(xin, Wl[l], bl[l], xl);
    k_edge_pass1<<<E / 32, 256, 0, stream>>>(xin, Rlayer, Wed[l], xl, att[l],
                                             srcS, dstS, etS, alphaS, amaxE);
    long long tot = (long long)E * D;
    k_edge_pass2<<<(int)((tot + 255) / 256), 256, 0, stream>>>(srcS, dstS, alphaS,
                                                               amaxE, xl, accum,
                                                               denom, E);
    k_finalize<<<(N * D + 255) / 256, 256, 0, stream>>>(accum, denom, bias[l],
                                                        l ? xB : xA, N, l ? 0 : 1);
  }

  k_head<<<BS / 16, 256, 0, stream>>>(xB, outw, outb, lng, lnb, (float*)d_out);
}